// EventVolumeSurface_46626164966023
// MI455X (gfx1250) — compile-verified
//
#include <hip/hip_runtime.h>
#include <hip/hip_bf16.h>
#include <stdint.h>

#define EV_H    480
#define EV_W    640
#define EV_BINS 10
#define EV_HW   (EV_H * EV_W)          /* 307200  */
#define EV_BHW  (EV_BINS * EV_HW)      /* 3072000 */
#define TILE    256

// ---------------------------------------------------------------------------
// Kernel 1: per-batch (t0, denom) table from lengths cumsum. B is tiny (8).
// ---------------------------------------------------------------------------
__global__ void ev_prep_kernel(const float* __restrict__ events,
                               const int* __restrict__ lengths,
                               int B, float2* __restrict__ tab) {
    if (blockIdx.x == 0 && threadIdx.x == 0) {
        int csum = 0;
        for (int b = 0; b < B; ++b) {
            int first = csum;
            csum += lengths[b];
            int last = csum - 1;
            float t0 = events[(size_t)first * 5 + 2];
            float tN = events[(size_t)last  * 5 + 2];
            tab[b] = make_float2(t0, tN - t0);
        }
    }
}

// ---------------------------------------------------------------------------
// Kernel 2: zero the output voxel grid (98 MB). Vectorized float4 stores.
// ---------------------------------------------------------------------------
__global__ void ev_zero_kernel(float4* __restrict__ out4, int n4) {
    int i = blockIdx.x * blockDim.x + threadIdx.x;
    if (i < n4) out4[i] = float4{0.f, 0.f, 0.f, 0.f};
}

// ---------------------------------------------------------------------------
// Kernel 3: main scatter. One block = one tile of 256 events.
// Stage the AoS event tile (1280 dwords) into LDS with CDNA5 async copies
// (fully coalesced b32 bursts), then read SoA from LDS (gcd(5,64)=1 ->
// bank-conflict-free) and issue up to 8 global_atomic_add_f32 per event.
// ---------------------------------------------------------------------------
__global__ void ev_scatter_kernel(const float* __restrict__ events,
                                  const float2* __restrict__ tab,
                                  float* __restrict__ out, int N) {
    __shared__ float smem[TILE * 5];

    const int tid  = threadIdx.x;
    const int base = blockIdx.x * TILE;       // first event index of tile
    const int dbase = base * 5;               // first dword of tile
    const int totalDwords = N * 5;

    const uint32_t lds_base = (uint32_t)(uintptr_t)smem;  // low 32b of generic = LDS addr

    // Async-stage 1280 contiguous dwords: thread t copies dwords
    // dbase + t + j*256 for j=0..4 -> each instruction is a coalesced burst.
#pragma unroll
    for (int j = 0; j < 5; ++j) {
        int d = dbase + tid + j * TILE;
        if (d < totalDwords) {
            uint32_t lds = lds_base + 4u * (uint32_t)(tid + j * TILE);
            uint64_t ga  = (uint64_t)(uintptr_t)(events + d);
            asm volatile("global_load_async_to_lds_b32 %0, %1, off"
                         :: "v"(lds), "v"(ga) : "memory");
        }
    }
    asm volatile("s_wait_asynccnt 0x0" ::: "memory");
    __syncthreads();

    const int idx = base + tid;
    if (idx >= N) return;

    const float x = smem[tid * 5 + 0];
    const float y = smem[tid * 5 + 1];
    const float t = smem[tid * 5 + 2];
    const float p = smem[tid * 5 + 3];
    const int  bi = (int)smem[tid * 5 + 4];

    const float2 tb = tab[bi];                // (t0, tN - t0)
    float u = (t - tb.x) / tb.y;              // same div-then-scale as reference
    u = fminf(fmaxf(u, 0.f), 1.f);
    const float ts = (float)(EV_BINS - 1) * u;

    const float xf = floorf(x), yf = floorf(y), bf = floorf(ts);
    const float fx = x - xf, fy = y - yf, ft = ts - bf;
    const int xi = (int)xf, yi = (int)yf, bzi = (int)bf;

    const float wx[2] = {1.f - fx, fx};
    const float wy[2] = {1.f - fy, fy};
    const float wt[2] = {1.f - ft, ft};

    const int batch_off = bi * EV_BHW;

#pragma unroll
    for (int dz = 0; dz < 2; ++dz) {
        const int bz = bzi + dz;
        // ceil corner invalid (reference masks it) exactly when ft==0 or OOB;
        // both produce a zero contribution -> skip the atomic entirely.
        if (dz && (ft == 0.f || bz >= EV_BINS)) continue;
        const float wz = p * wt[dz];
#pragma unroll
        for (int dy = 0; dy < 2; ++dy) {
            const int yy = yi + dy;
            if (dy && (fy == 0.f || yy >= EV_H)) continue;
            const float wzy = wz * wy[dy];
            const int row_off = batch_off + bz * EV_HW + yy * EV_W;
#pragma unroll
            for (int dx = 0; dx < 2; ++dx) {
                const int xx = xi + dx;
                if (dx && (fx == 0.f || xx >= EV_W)) continue;
                const float w = wzy * wx[dx];
                atomicAdd(&out[row_off + xx], w);   // global_atomic_add_f32 (no return)
            }
        }
    }
}

// ---------------------------------------------------------------------------
// Launcher
// ---------------------------------------------------------------------------
extern "C" void kernel_launch(void* const* d_in, const int* in_sizes, int n_in,
                              void* d_out, int out_size, void* d_ws, size_t ws_size,
                              hipStream_t stream) {
    const float* events  = (const float*)d_in[0];   // [N,5] f32
    const int*   lengths = (const int*)d_in[1];     // [B]   i32
    float*       out     = (float*)d_out;           // [B,BINS,H,W] f32

    const int N = in_sizes[0] / 5;
    const int B = in_sizes[1];

    float2* tab = (float2*)d_ws;                    // 8 * 8 bytes of scratch

    // 1) per-batch (t0, denom)
    ev_prep_kernel<<<1, 32, 0, stream>>>(events, lengths, B, tab);

    // 2) zero output (out_size = 24,576,000, divisible by 4)
    const int n4 = out_size / 4;
    ev_zero_kernel<<<(n4 + 255) / 256, 256, 0, stream>>>((float4*)out, n4);

    // 3) scatter
    const int ntiles = (N + TILE - 1) / TILE;
    ev_scatter_kernel<<<ntiles, TILE, 0, stream>>>(events, tab, out, N);
}